// AttentionHead_17600775979058
// MI455X (gfx1250) — compile-verified
//
#include <hip/hip_runtime.h>
#include <hip/hip_bf16.h>
#include <math.h>

typedef __attribute__((ext_vector_type(2))) float v2f;
typedef __attribute__((ext_vector_type(8))) float v8f;

#define NB 16
#define NM 50
#define FMS 256
#define CHAN_ELEMS (FMS * FMS)          // 65536
#define MASK_ELEMS ((size_t)NB * NM * CHAN_ELEMS)

// ---------------------------------------------------------------------------
// Kernel 1: per-(b,n) separable axes. exs carries factor1 folded in.
// Also block-reduce min/max of each axis -> per-(b,n) mask min/max.
// grid = 800 blocks, block = 256 threads (one thread per grid point).
// ---------------------------------------------------------------------------
__global__ void build_axes(const float* __restrict__ boxes,
                           float* __restrict__ exs,
                           float* __restrict__ eys,
                           float* __restrict__ pmm) {
    const int bn = blockIdx.x;        // 0..799  == b*50+n
    const int i  = threadIdx.x;       // 0..255  grid index

    const float x = boxes[bn * 4 + 0];
    const float y = boxes[bn * 4 + 1];
    const float w = boxes[bn * 4 + 2];
    const float h = boxes[bn * 4 + 3];

    const float xc = x + floorf(w * 0.5f);
    const float yc = y + floorf(h * 0.5f);
    // gx = round(linspace(0, 256, 256)) -> round(i * 256/255)
    const float g = rintf((float)i * (256.0f / 255.0f));
    // factor1 = (2*pi)^-1 * (0.0625*w*h)^-0.5
    const float f1 = 0.15915494309189535f / sqrtf(0.0625f * w * h);

    const float dx = g - xc;
    const float dy = g - yc;
    const float ex = f1 * __expf(-0.5f * dx * dx / (0.25f * w));
    const float ey =      __expf(-0.5f * dy * dy / (0.25f * h));

    exs[bn * FMS + i] = ex;
    eys[bn * FMS + i] = ey;

    // block tree-reduce: min/max of ex and ey (all values > 0)
    __shared__ float s0[256], s1[256], s2[256], s3[256];
    s0[i] = ex; s1[i] = ex; s2[i] = ey; s3[i] = ey;
    __syncthreads();
    for (int s = 128; s > 0; s >>= 1) {
        if (i < s) {
            s0[i] = fminf(s0[i], s0[i + s]);
            s1[i] = fmaxf(s1[i], s1[i + s]);
            s2[i] = fminf(s2[i], s2[i + s]);
            s3[i] = fmaxf(s3[i], s3[i + s]);
        }
        __syncthreads();
    }
    if (i == 0) {
        pmm[2 * bn + 0] = s0[0] * s2[0];   // mask min for this (b,n)
        pmm[2 * bn + 1] = s1[0] * s3[0];   // mask max
    }
}

// ---------------------------------------------------------------------------
// Kernel 2: reduce 50 per-mask extrema per batch -> mn[b], scale[b]=50/(mx-mn)
// ---------------------------------------------------------------------------
__global__ void reduce_batch(const float* __restrict__ pmm,
                             float* __restrict__ bstats) {
    const int b = threadIdx.x;
    if (b >= NB) return;
    float mn = 3.402823466e38f, mx = -3.402823466e38f;
    for (int n = 0; n < NM; ++n) {
        mn = fminf(mn, pmm[2 * (b * NM + n) + 0]);
        mx = fmaxf(mx, pmm[2 * (b * NM + n) + 1]);
    }
    bstats[2 * b + 0] = mn;
    bstats[2 * b + 1] = 50.0f / (mx - mn);
}

// ---------------------------------------------------------------------------
// Kernel 3: permuted label copy (800 values at tail of d_out)
// ---------------------------------------------------------------------------
__global__ void copy_labels(const float* __restrict__ labels,
                            float* __restrict__ out_labels) {
    const int idx = blockIdx.x * blockDim.x + threadIdx.x;
    if (idx >= NB * NM) return;
    const int b = idx / NM;
    const int c = idx % NM;
    const int n = (c < 25) ? (2 * c) : (2 * (c - 25) + 1);   // channel shuffle
    out_labels[idx] = labels[b * NM + n];
}

// ---------------------------------------------------------------------------
// Kernel 4: main streaming kernel. One block per output channel (b,c).
// ex/ey rows staged in LDS once per block. Each wave owns tile-rows
// ti = wave and ti = wave+8; A fragment fixed per pass, all 16 B fragments
// preloaded into registers. Fully unrolled: 32 v_wmma_f32_16x16x4_f32 per
// wave, every store an immediate-offset from one per-pass base (no EXEC
// divergence anywhere -> WMMA's EXEC==all-ones requirement holds).
//   D = (scale*ex) (x) ey + (-mn*scale)  -> final normalized output tile.
// ---------------------------------------------------------------------------
__global__ void outer_wmma(const float* __restrict__ exs,
                           const float* __restrict__ eys,
                           const float* __restrict__ bstats,
                           float* __restrict__ out) {
    const int blk = blockIdx.x;            // 0..799  == b*50 + c (output order)
    const int b = blk / NM;
    const int c = blk % NM;
    const int n = (c < 25) ? (2 * c) : (2 * (c - 25) + 1);   // source mask

    const float mn = bstats[2 * b + 0];
    const float sc = bstats[2 * b + 1];
    const float bias = -mn * sc;

    __shared__ float sex[FMS];
    __shared__ float sey[FMS];
    {
        const int t = threadIdx.x;         // 0..255
        sex[t] = exs[(b * NM + n) * FMS + t] * sc;   // fold scale into A
        sey[t] = eys[(b * NM + n) * FMS + t];
    }
    __syncthreads();

    float* __restrict__ chan = out + (size_t)blk * CHAN_ELEMS;

    const int lane = threadIdx.x & 31;
    const int wave = threadIdx.x >> 5;     // 0..7
    const int l15  = lane & 15;
    const bool lo  = (lane < 16);

    // Preload all 16 B fragments (4x16 f32, K=0 row = ey chunk, rest 0).
    // Unconditional LDS load for every lane + cndmask: no EXEC branching.
    v2f Bm[16];
    #pragma unroll
    for (int tj = 0; tj < 16; ++tj) {
        const float v = sey[tj * 16 + l15];
        Bm[tj].x = lo ? v : 0.0f;
        Bm[tj].y = 0.0f;
    }

    #pragma unroll
    for (int p = 0; p < 2; ++p) {
        const int ti = wave + (p << 3);    // tile row: 0..15
        const float va = sex[ti * 16 + l15];
        v2f A;
        A.x = lo ? va : 0.0f;              // 16x4 f32 A: K=0 column only
        A.y = 0.0f;

        // per-pass store base: row = ti*16 + 8*(lane>=16) (+r), col = l15 (+tj*16)
        float* __restrict__ base =
            chan + (ti * 16 + ((lane >> 4) << 3)) * FMS + l15;

        #pragma unroll
        for (int tj = 0; tj < 16; ++tj) {
            v8f C;
            #pragma unroll
            for (int r = 0; r < 8; ++r) C[r] = bias;

            v8f D = __builtin_amdgcn_wmma_f32_16x16x4_f32(
                /*neg_a=*/false, A, /*neg_b=*/false, Bm[tj],
                /*c_mod=*/(short)0, C, /*reuse_a=*/false, /*reuse_b=*/false);

            // D: VGPR r, lane l -> M = r + 8*(l>=16), N = l%16
            #pragma unroll
            for (int r = 0; r < 8; ++r)
                base[r * FMS + tj * 16] = D[r];   // imm offset: r*1024 + tj*64 B
        }
    }
}

// ---------------------------------------------------------------------------
extern "C" void kernel_launch(void* const* d_in, const int* in_sizes, int n_in,
                              void* d_out, int out_size, void* d_ws, size_t ws_size,
                              hipStream_t stream) {
    const float* boxes  = (const float*)d_in[0];   // (16,50,4)
    const float* labels = (const float*)d_in[1];   // (16,50,1)
    float* out = (float*)d_out;

    // workspace layout (floats)
    float* ws     = (float*)d_ws;
    float* exs    = ws;                    // 800*256
    float* eys    = exs + NB * NM * FMS;   // 800*256
    float* pmm    = eys + NB * NM * FMS;   // 800*2
    float* bstats = pmm + NB * NM * 2;     // 16*2

    build_axes<<<NB * NM, 256, 0, stream>>>(boxes, exs, eys, pmm);
    reduce_batch<<<1, 32, 0, stream>>>(pmm, bstats);
    copy_labels<<<(NB * NM + 255) / 256, 256, 0, stream>>>(labels, out + MASK_ELEMS);
    outer_wmma<<<NB * NM, 256, 0, stream>>>(exs, eys, bstats, out);
}